// OnnxNonMaxSuppression_6021544149739
// MI455X (gfx1250) — compile-verified
//
#include <hip/hip_runtime.h>
#include <stdint.h>

// ---------------------------------------------------------------------------
// ONNX NonMaxSuppression for MI455X (gfx1250, wave32).
//   boxes  : [B, N, 4] f32 (x1,y1,x2,y2)
//   scores : [B, C, N] f32
//   out    : [B*C*max_out, 3] int64 triplets (b, c, idx) padded with -1
// One workgroup per (b,c). Scores held in VGPRs (32/thread), boxes staged
// into LDS once via the Tensor Data Mover, greedy loop does wave32-shuffle
// argmax + register suppression.
// ---------------------------------------------------------------------------

#define NMS_B   8
#define NMS_C   80
#define NMS_N   8192
#define THREADS 256
#define ITEMS   (NMS_N / THREADS)   // 32 scores per thread, kept in VGPRs
#define NWAVES  (THREADS / 32)      // 8 waves (wave32)

typedef __attribute__((ext_vector_type(4))) unsigned int tdm_u32x4;
typedef __attribute__((ext_vector_type(8))) int          tdm_i32x8;
typedef __attribute__((ext_vector_type(4))) int          tdm_i32x4;

__global__ __launch_bounds__(THREADS)
void nms_gfx1250_kernel(const float* __restrict__ boxes,
                        const float* __restrict__ scores,
                        const float* __restrict__ iou_thr_p,
                        const float* __restrict__ score_thr_p,
                        long long*   __restrict__ out,
                        int max_out)
{
    __shared__ float4 sBoxes[NMS_N];      // 128 KB: all boxes of this batch
    __shared__ float  sRedV[NWAVES];
    __shared__ int    sRedI[NWAVES];
    __shared__ float  sSelV;
    __shared__ int    sSelI;

    const int bc   = blockIdx.x;          // 0 .. B*C-1
    const int b    = bc / NMS_C;
    const int c    = bc % NMS_C;
    const int t    = threadIdx.x;
    const int wave = t >> 5;
    const int lane = t & 31;

    const float iouThr   = iou_thr_p[0];
    const float scoreThr = score_thr_p[0];
    const float NEG_INF  = -__builtin_inff();

    // ---- Stage boxes[b] (N x 4 f32 = 128 KB) into LDS ----------------------
#if __has_builtin(__builtin_amdgcn_tensor_load_to_lds)
    if (t < 32) {   // one wave issues the TDM DMA; TENSORcnt is per-wave
        const unsigned long long gaddr =
            (unsigned long long)(uintptr_t)(boxes + (size_t)b * NMS_N * 4);
        const unsigned int laddr  = (unsigned int)(uintptr_t)(&sBoxes[0]);
        const unsigned int nwords = NMS_N * 4u;           // 32768 f32 elements

        tdm_u32x4 g0;
        g0[0] = 1u;                                        // count=1 (valid), user mode
        g0[1] = laddr;                                     // lds_addr  (bits 63:32)
        g0[2] = (unsigned int)(gaddr & 0xFFFFFFFFull);     // global_addr lo
        g0[3] = (unsigned int)((gaddr >> 32) & 0x1FFFFFFull)
              | (2u << 30);                                // global_addr hi + type=2

        tdm_i32x8 g1;
        g1[0] = (int)(2u << 16);                           // data_size=2 -> 4 bytes
        g1[1] = (int)((nwords & 0xFFFFu) << 16);           // tensor_dim0[15:0]
        g1[2] = (int)(((nwords >> 16) & 0xFFFFu)           // tensor_dim0[31:16]
              | (1u << 16));                               // tensor_dim1 = 1
        g1[3] = (int)((nwords & 0xFFFFu) << 16);           // tile_dim0 = nwords
        g1[4] = 1;                                         // tile_dim1=1, tile_dim2=0
        g1[5] = (int)nwords;                               // tensor_dim0_stride lo32
        g1[6] = 0;                                         // stride hi / dim1_stride
        g1[7] = 0;

        tdm_i32x4 g2 = {0, 0, 0, 0};
        tdm_i32x4 g3 = {0, 0, 0, 0};
#if __clang_major__ >= 23
        tdm_i32x8 g4 = {0, 0, 0, 0, 0, 0, 0, 0};
        __builtin_amdgcn_tensor_load_to_lds(g0, g1, g2, g3, g4, 0);
#else
        __builtin_amdgcn_tensor_load_to_lds(g0, g1, g2, g3, 0);
#endif
        __builtin_amdgcn_s_wait_tensorcnt(0);
    }
#else
    // Fallback: cooperative float4 copy (also keeps host pass compiling).
    for (int i = t; i < NMS_N; i += THREADS)
        sBoxes[i] = ((const float4*)boxes)[(size_t)b * NMS_N + i];
#endif
    __syncthreads();

    // ---- Load this class's scores into registers, apply score threshold ----
    const float* sc = scores + (size_t)bc * NMS_N;
    float s[ITEMS];
#pragma unroll
    for (int e = 0; e < ITEMS; ++e) {
        float v = sc[e * THREADS + t];          // coalesced: stride-256 per e
        s[e] = (v > scoreThr) ? v : NEG_INF;
    }

    long long* outBase = out + (size_t)bc * (size_t)max_out * 3;

    // ---- Greedy NMS loop ---------------------------------------------------
    int k = 0;
    for (; k < max_out; ++k) {
        // Per-thread argmax over 32 register scores (ascending index => ties
        // naturally resolve to the lowest index within a thread).
        float bv = NEG_INF;
        int   bi = 0x7FFFFFFF;
#pragma unroll
        for (int e = 0; e < ITEMS; ++e) {
            int i = e * THREADS + t;
            if (s[e] > bv) { bv = s[e]; bi = i; }
        }
        // Wave32 argmax via xor-shuffle tree (5 steps), ties -> lower index.
#pragma unroll
        for (int m = 16; m >= 1; m >>= 1) {
            float ov = __shfl_xor(bv, m, 32);
            int   oi = __shfl_xor(bi, m, 32);
            if (ov > bv || (ov == bv && oi < bi)) { bv = ov; bi = oi; }
        }
        if (lane == 0) { sRedV[wave] = bv; sRedI[wave] = bi; }
        __syncthreads();
        if (t == 0) {
            float gv = sRedV[0]; int gi = sRedI[0];
#pragma unroll
            for (int w = 1; w < NWAVES; ++w) {
                float wv = sRedV[w]; int wi = sRedI[w];
                if (wv > gv || (wv == gv && wi < gi)) { gv = wv; gi = wi; }
            }
            sSelV = gv; sSelI = gi;
        }
        __syncthreads();

        const float selV = sSelV;               // LDS broadcast -> uniform
        const int   sel  = sSelI;
        if (!(selV > NEG_INF)) break;           // nothing above threshold left

        if (t == 0) {
            outBase[(size_t)k * 3 + 0] = (long long)b;
            outBase[(size_t)k * 3 + 1] = (long long)c;
            outBase[(size_t)k * 3 + 2] = (long long)sel;
        }

        // Suppress: IoU of selected box vs all boxes; scores stay in VGPRs.
        const float4 sb = sBoxes[sel];          // same-address LDS broadcast
        const float selArea =
            fmaxf(sb.z - sb.x, 0.f) * fmaxf(sb.w - sb.y, 0.f);
#pragma unroll
        for (int e = 0; e < ITEMS; ++e) {
            int i = e * THREADS + t;
            float4 o = sBoxes[i];               // ds_load_b128, conflict-free
            float iw = fminf(o.z, sb.z) - fmaxf(o.x, sb.x);
            float ih = fminf(o.w, sb.w) - fmaxf(o.y, sb.y);
            iw = fmaxf(iw, 0.f);
            ih = fmaxf(ih, 0.f);
            float inter = iw * ih;
            float oarea = fmaxf(o.z - o.x, 0.f) * fmaxf(o.w - o.y, 0.f);
            float uni   = oarea + selArea - inter;
            float iou   = (uni > 0.f) ? (inter / uni) : 0.f;
            if (iou > iouThr || i == sel) s[e] = NEG_INF;
        }
    }

    // ---- Pad remaining slots with (-1,-1,-1), all threads in parallel ------
    for (int x = k * 3 + t; x < max_out * 3; x += THREADS)
        outBase[x] = -1LL;
}

// ---------------------------------------------------------------------------
extern "C" void kernel_launch(void* const* d_in, const int* in_sizes, int n_in,
                              void* d_out, int out_size, void* d_ws, size_t ws_size,
                              hipStream_t stream)
{
    (void)in_sizes; (void)n_in; (void)d_ws; (void)ws_size;

    const float* boxes     = (const float*)d_in[0];   // [B,N,4] f32
    const float* scores    = (const float*)d_in[1];   // [B,C,N] f32
    // d_in[2] = max_output_boxes_per_class (device int) -- derived from
    // out_size instead, since we cannot do a sync D2H copy under capture.
    const float* iou_thr   = (const float*)d_in[3];   // [1] f32
    const float* score_thr = (const float*)d_in[4];   // [1] f32
    long long*   out       = (long long*)d_out;       // [B*C*max_out, 3] i64

    const int max_out = out_size / (NMS_B * NMS_C * 3);
    if (max_out <= 0) return;

    nms_gfx1250_kernel<<<dim3(NMS_B * NMS_C), dim3(THREADS), 0, stream>>>(
        boxes, scores, iou_thr, score_thr, out, max_out);
}